// EntropyLoss_35948876267666
// MI455X (gfx1250) — compile-verified
//
#include <hip/hip_runtime.h>
#include <math.h>

// Problem geometry (fixed by the reference: heatmap (8, 20, 512, 512) fp32)
#define NBATCH 8
#define NCHAN  20
#define HWDIM  (512 * 512)          // 262144 elements per channel
#define SLICES 8                    // blocks per channel
#define SLICE_ELEMS (HWDIM / SLICES)        // 32768
#define BLOCK_THREADS 256
#define WAVES 8                              // 256 / wave32
#define WAVE_ELEMS (SLICE_ELEMS / WAVES)     // 4096 floats per wave
#define TILE_ELEMS 512                       // floats per LDS buffer per wave
#define NTILES (WAVE_ELEMS / TILE_ELEMS)     // 8 tiles, double buffered
#define CHUNKS 4                             // async b128 issues per tile (4*32*4 = 512 floats)

#define LN2F 0.69314718055994530942f

typedef float v2f __attribute__((ext_vector_type(2)));
typedef float v8f __attribute__((ext_vector_type(8)));

// Full-wave32 sum via V_WMMA_F32_16X16X4_F32.
// A: lanes 0-15 hold (M=lane, K=0) in VGPR0, K=1 in VGPR1; lanes 16-31 hold K=2,3.
// We place x in K=0/2 (a.x) and 0 in K=1/3 (a.y). B = all ones (layout independent).
// D[M][*] = x[M] + x[M+16]. Lane j<16 holds rows 0..7 in its 8 D VGPRs, lane j>=16
// holds rows 8..15 -> sum the 8 VGPRs, then one xor-16 shuffle completes the sum.
__device__ __forceinline__ float wave_sum_wmma(float x) {
    v2f a; a.x = x;    a.y = 0.0f;
    v2f b; b.x = 1.0f; b.y = 1.0f;
    v8f c = {};
    v8f d = __builtin_amdgcn_wmma_f32_16x16x4_f32(
        /*neg_a=*/false, a, /*neg_b=*/false, b,
        /*c_mod=*/(short)0, c, /*reuse_a=*/false, /*reuse_b=*/false);
    float r = d[0] + d[1] + d[2] + d[3] + d[4] + d[5] + d[6] + d[7];
    r += __shfl_xor(r, 16, 32);
    return r;
}

// Issue one 512-float tile as 4 async global->LDS b128 copies (per-lane 16B).
__device__ __forceinline__ void issue_tile_async(const float* gsrc,
                                                 unsigned lds_base_bytes,
                                                 int lane) {
#pragma unroll
    for (int q = 0; q < CHUNKS; ++q) {
        unsigned lds_addr = lds_base_bytes + (unsigned)((q * 128 + lane * 4) * 4);
        unsigned long long gaddr =
            (unsigned long long)(const void*)(gsrc + q * 128 + lane * 4);
        asm volatile("global_load_async_to_lds_b128 %0, %1, off"
                     :: "v"(lds_addr), "v"(gaddr)
                     : "memory");
    }
}

// Stage 1: per-(channel-slice) partial sums S = sum(e^x), T = sum(x e^x),
// CNT = sum(x>0), streamed through a double-buffered async LDS pipeline.
__global__ void __launch_bounds__(BLOCK_THREADS)
entropy_partials_kernel(const float* __restrict__ heat, float4* __restrict__ partials) {
    __shared__ float tile[WAVES * 2 * TILE_ELEMS];   // 32 KB
    __shared__ float wsum[WAVES][4];

    const int lane = threadIdx.x & 31;
    const int wave = threadIdx.x >> 5;
    const int ch    = blockIdx.x / SLICES;           // n*NCHAN + c
    const int slice = blockIdx.x % SLICES;

    const float* base = heat + (size_t)ch * HWDIM
                             + (size_t)slice * SLICE_ELEMS
                             + (size_t)wave * WAVE_ELEMS;
    const unsigned ldsWaveBase =
        (unsigned)(uintptr_t)(&tile[0]) + (unsigned)(wave * 2 * TILE_ELEMS * 4);

    float s = 0.0f, t = 0.0f, cnt = 0.0f;

    // Prologue: prefetch tile 0 into buffer 0.
    issue_tile_async(base, ldsWaveBase, lane);

    for (int tl = 0; tl < NTILES; ++tl) {
        const int buf = tl & 1;
        if (tl + 1 < NTILES) {
            issue_tile_async(base + (tl + 1) * TILE_ELEMS,
                             ldsWaveBase + (unsigned)(((tl + 1) & 1) * TILE_ELEMS * 4),
                             lane);
            // 4 newly-issued remain in flight; previous 4 (this tile) are done.
            asm volatile("s_wait_asynccnt 0x4" ::: "memory");
        } else {
            asm volatile("s_wait_asynccnt 0x0" ::: "memory");
        }

        const float4* tp = reinterpret_cast<const float4*>(
            &tile[(wave * 2 + buf) * TILE_ELEMS]);
#pragma unroll
        for (int q = 0; q < CHUNKS; ++q) {
            float4 v = tp[q * 32 + lane];
            {
                float x = v.x; float e = x > 0.0f ? __expf(x) : 0.0f;
                s += e; t += x * e; cnt += x > 0.0f ? 1.0f : 0.0f;
            }
            {
                float x = v.y; float e = x > 0.0f ? __expf(x) : 0.0f;
                s += e; t += x * e; cnt += x > 0.0f ? 1.0f : 0.0f;
            }
            {
                float x = v.z; float e = x > 0.0f ? __expf(x) : 0.0f;
                s += e; t += x * e; cnt += x > 0.0f ? 1.0f : 0.0f;
            }
            {
                float x = v.w; float e = x > 0.0f ? __expf(x) : 0.0f;
                s += e; t += x * e; cnt += x > 0.0f ? 1.0f : 0.0f;
            }
        }
    }

    // Wave32 reduction via WMMA (EXEC is all ones: no divergent control flow).
    float ws_ = wave_sum_wmma(s);
    float wt_ = wave_sum_wmma(t);
    float wc_ = wave_sum_wmma(cnt);
    if (lane == 0) {
        wsum[wave][0] = ws_;
        wsum[wave][1] = wt_;
        wsum[wave][2] = wc_;
    }
    __syncthreads();

    if (threadIdx.x == 0) {
        float S = 0.0f, T = 0.0f, C = 0.0f;
#pragma unroll
        for (int w = 0; w < WAVES; ++w) {
            S += wsum[w][0]; T += wsum[w][1]; C += wsum[w][2];
        }
        partials[blockIdx.x] = make_float4(S, T, C, 0.0f);
    }
}

// Stage 2: one wave per batch image; fold slice partials per channel,
// ent_c = ln(S) - T/S (shift-invariant entropy in nats), then reduce channels.
__global__ void __launch_bounds__(32)
entropy_final_kernel(const float4* __restrict__ partials, float* __restrict__ out) {
    const int n = blockIdx.x;
    const int lane = threadIdx.x;

    float ent = 0.0f, cnt = 0.0f;
    for (int c = lane; c < NCHAN; c += 32) {
        float S = 0.0f, T = 0.0f, C = 0.0f;
#pragma unroll
        for (int b = 0; b < SLICES; ++b) {
            float4 p = partials[(n * NCHAN + c) * SLICES + b];
            S += p.x; T += p.y; C += p.z;
        }
        if (S > 0.0f) ent += (logf(S) - T / S);
        cnt += C;
    }
#pragma unroll
    for (int off = 16; off > 0; off >>= 1) {
        ent += __shfl_down(ent, off, 32);
        cnt += __shfl_down(cnt, off, 32);
    }
    if (lane == 0) out[n] = ent / (LN2F * cnt);
}

extern "C" void kernel_launch(void* const* d_in, const int* in_sizes, int n_in,
                              void* d_out, int out_size, void* d_ws, size_t ws_size,
                              hipStream_t stream) {
    (void)in_sizes; (void)n_in; (void)out_size; (void)ws_size;
    const float* heat = (const float*)d_in[0];
    float* out = (float*)d_out;
    float4* partials = (float4*)d_ws;    // NBATCH*NCHAN*SLICES = 1280 float4 = 20 KB

    entropy_partials_kernel<<<dim3(NBATCH * NCHAN * SLICES), dim3(BLOCK_THREADS),
                              0, stream>>>(heat, partials);
    entropy_final_kernel<<<dim3(NBATCH), dim3(32), 0, stream>>>(partials, out);
}